// RW_layer_44289702756536
// MI455X (gfx1250) — compile-verified
//
#include <hip/hip_runtime.h>

// ---------------------------------------------------------------------------
// RW graph-filter layer for MI455X (gfx1250, wave32, WMMA).
// Precision: f32 throughout via V_WMMA_F32_16X16X4_F32 (compute is ~41 GFLOP
// total vs ~200MB of mandatory HBM traffic at 23.3 TB/s -> memory bound, so
// f32 matrix cores are the right choice; no quantization needed).
// ---------------------------------------------------------------------------

typedef float v2f __attribute__((ext_vector_type(2)));
typedef float v8f __attribute__((ext_vector_type(8)));

#define B_BATCH 1024
#define NNODE   128
#define CDIM    128
#define MHID    10
#define OUTD    32
#define NPAD    136   // 128 + 8 pad: row stride mod 64 banks = 8 dwords -> conflict-free A/B frag reads

// D = A(16x4) * B(4x16) + C, all f32.
// A frag: lane l holds row (l&15), k = kit*4 + (l>>4)*2 (+1)   -> v2f
// B frag: lane l holds col (l&15), k = kit*4 + (l>>4)*2 (+1)   -> v2f
// C/D:    lane l holds col (l&15), rows r + 8*(l>>4), r=0..7   -> v8f
__device__ __forceinline__ v8f wmma4(v2f a, v2f b, v8f c) {
  return __builtin_amdgcn_wmma_f32_16x16x4_f32(
      /*neg_a=*/false, a, /*neg_b=*/false, b,
      /*c_mod=*/(short)0, c, /*reuse_a=*/false, /*reuse_b=*/false);
}

// Workspace layout (floats):
//   [0      .. 3200)    A dense [10][10][32]
//   [3200   .. 44160)   z1 [10][128][32]
//   [44160  .. 85120)   z2 [10][128][32]
//   [85120  .. 208000)  zfrag[3][10][2][32][32] of float2 (B-operand fragments)
#define WS_A   0
#define WS_Z1  3200
#define WS_Z2  44160
#define WS_ZF  85120
#define ZF_STRIDE 40960   // floats per step table (10*2*32*32*2)

// ---------------------------------------------------------------------------
// Pre-kernel: batch-invariant hidden-graph math + B-fragment repack.
// ---------------------------------------------------------------------------
__global__ __launch_bounds__(256) void rw_pre(const float* __restrict__ fh,   // [10][128][32]
                                              const float* __restrict__ ah,   // [45][32]
                                              float* __restrict__ ws) {
  float* Adense = ws + WS_A;
  float* z1     = ws + WS_Z1;
  float* z2     = ws + WS_Z2;
  float* zf     = ws + WS_ZF;
  const int tid = threadIdx.x;

  // A[i][j][d] = relu(ah[p(i,j)][d]) symmetric, zero diagonal.
  for (int e = tid; e < MHID * MHID * OUTD; e += 256) {
    int d = e & 31, ij = e >> 5, i2 = ij / MHID, j2 = ij % MHID;
    float v = 0.f;
    if (i2 != j2) {
      int i = i2 < j2 ? i2 : j2;
      int j = i2 < j2 ? j2 : i2;
      int p = i * MHID - (i * (i + 1)) / 2 + (j - i - 1);
      float t = ah[p * OUTD + d];
      v = t > 0.f ? t : 0.f;
    }
    Adense[e] = v;
  }
  __syncthreads();

  // z1[a][c][d] = sum_b A[a][b][d] * z0[b][c][d]
  for (int e = tid; e < MHID * CDIM * OUTD; e += 256) {
    int d = e & 31, c = (e >> 5) & 127, a = e >> 12;
    float s = 0.f;
#pragma unroll
    for (int bb = 0; bb < MHID; ++bb)
      s += Adense[(a * MHID + bb) * OUTD + d] * fh[(bb * CDIM + c) * OUTD + d];
    z1[e] = s;
  }
  __syncthreads();

  // z2 = A . z1
  for (int e = tid; e < MHID * CDIM * OUTD; e += 256) {
    int d = e & 31, c = (e >> 5) & 127, a = e >> 12;
    float s = 0.f;
#pragma unroll
    for (int bb = 0; bb < MHID; ++bb)
      s += Adense[(a * MHID + bb) * OUTD + d] * z1[((bb * CDIM + c) << 5) + d];
    z2[e] = s;
  }
  __syncthreads();

  // Repack z0/z1/z2 into WMMA B fragments:
  // zfrag[s][m][ot][kit][lane] = { Z[m][k][o], Z[m][k+1][o] },
  //   k = kit*4 + (lane>>4)*2, o = ot*16 + (lane&15)
  for (int s = 0; s < 3; ++s) {
    const float* zs = (s == 0) ? fh : (s == 1 ? z1 : z2);
    float* dst = zf + s * ZF_STRIDE;
    for (int e = tid; e < MHID * 2 * 32 * 32; e += 256) {
      int lane = e & 31, kit = (e >> 5) & 31, ot = (e >> 10) & 1, m = e >> 11;
      int k = kit * 4 + (lane >> 4) * 2;
      int o = ot * 16 + (lane & 15);
      dst[e * 2 + 0] = zs[(m * CDIM + k) * OUTD + o];
      dst[e * 2 + 1] = zs[(m * CDIM + k + 1) * OUTD + o];
    }
  }
}

// ---------------------------------------------------------------------------
// Main kernel: one workgroup (8 wave32) per batch.
// ---------------------------------------------------------------------------
__global__ __launch_bounds__(256) void rw_main(const float* __restrict__ adj,   // [B][128][128]
                                               const float* __restrict__ feats, // [N][128]
                                               const int*   __restrict__ idxs,  // [B][128]
                                               const float* __restrict__ zfrag, // [3][...] frag tables
                                               float*       __restrict__ out) { // [B][32]
  __shared__ float lx0[NNODE * NPAD];  // gathered X0
  __shared__ float lxc[NNODE * NPAD];  // propagated X_s (s>=1)
  __shared__ float accbuf[OUTD];

  const int b   = blockIdx.x;
  const int tid = threadIdx.x;
  const int w   = tid >> 5;
  const int l   = tid & 31;
  const int l16 = l & 15;
  const int lhi = l >> 4;

  if (tid < OUTD) accbuf[tid] = 0.f;

  // ---- gather X0 rows (16 rows per wave, float4 per lane, coalesced) ----
  {
    const int r0 = w * 16;
#pragma unroll 4
    for (int r = r0; r < r0 + 16; ++r) {
      int idx = idxs[b * NNODE + r];
      float4 v = *(const float4*)(feats + (size_t)idx * CDIM + l * 4);
      *(float4*)(&lx0[r * NPAD + l * 4]) = v;
    }
  }
  __syncthreads();

  float acc0 = 0.f, acc1 = 0.f;          // per-lane partials for out tiles 0/1
  const float* xcur = lx0;

  for (int step = 0; step < 3; ++step) {
    if (step) {
      // ---- X_next = adj_b @ X_cur : wave w owns column tile ct=w, all 8 row tiles.
      // Accumulators stay in registers; write back to lxc after a full barrier.
      v8f cacc[8];
      const float* Ag = adj + (size_t)b * NNODE * CDIM;
      const int ct = w;
      const int col = ct * 16 + l16;
#pragma unroll
      for (int nt = 0; nt < 8; ++nt) {
        v8f c = {};
        const int row = nt * 16 + l16;
#pragma unroll 4
        for (int kit = 0; kit < 32; ++kit) {
          const int kb = kit * 4 + lhi * 2;
          v2f a = *(const v2f*)(Ag + row * CDIM + kb);     // global_load_b64
          v2f bv;
          bv.x = xcur[(kb + 0) * NPAD + col];              // ds_load_b32
          bv.y = xcur[(kb + 1) * NPAD + col];
          c = wmma4(a, bv, c);
        }
        cacc[nt] = c;
      }
      __syncthreads();   // everyone done reading xcur
#pragma unroll
      for (int nt = 0; nt < 8; ++nt)
#pragma unroll
        for (int r = 0; r < 8; ++r)
          lxc[(nt * 16 + r + lhi * 8) * NPAD + ct * 16 + l16] = cacc[nt][r];
      __syncthreads();   // X_next visible
      xcur = lxc;
    }

    // ---- contribution pass: tiles (m, nt, ot), 160 tiles / 8 waves.
    // czx uses (z0 frags, X0); ct uses (z_step frags, X_step). step==0 -> ct = czx.
    const float* zf0 = zfrag;
    const float* zfi = zfrag + step * ZF_STRIDE;
    for (int t = w; t < MHID * 8 * 2; t += 8) {
      const int m  = t >> 4;
      const int nt = (t >> 1) & 7;
      const int ot = t & 1;
      const int row = nt * 16 + l16;
      const v2f* bz0 = (const v2f*)zf0 + (size_t)(m * 2 + ot) * 1024 + l;
      const v2f* bzi = (const v2f*)zfi + (size_t)(m * 2 + ot) * 1024 + l;
      v8f czx = {}, ctt = {};
#pragma unroll 4
      for (int kit = 0; kit < 32; ++kit) {
        const int kb = kit * 4 + lhi * 2;
        v2f a0 = *(const v2f*)(&lx0[row * NPAD + kb]);     // ds_load_b64
        czx = wmma4(a0, bz0[kit * 32], czx);               // global_load_b64 (coalesced, L2-hot)
        if (step) {
          v2f ai = *(const v2f*)(&xcur[row * NPAD + kb]);
          ctt = wmma4(ai, bzi[kit * 32], ctt);
        }
      }
      if (!step) ctt = czx;
      float s = 0.f;
#pragma unroll
      for (int r = 0; r < 8; ++r) s += czx[r] * ctt[r];
      if (ot == 0) acc0 += s; else acc1 += s;
    }
  }

  // ---- reduce across lanes/waves into LDS, scale, store ----
  atomicAdd(&accbuf[l16], acc0);          // ds_add_f32
  atomicAdd(&accbuf[16 + l16], acc1);
  __syncthreads();
  if (tid < OUTD) out[(size_t)b * OUTD + tid] = accbuf[tid] * (1.0f / 3840.0f);
}

// ---------------------------------------------------------------------------
extern "C" void kernel_launch(void* const* d_in, const int* in_sizes, int n_in,
                              void* d_out, int out_size, void* d_ws, size_t ws_size,
                              hipStream_t stream) {
  (void)in_sizes; (void)n_in; (void)out_size; (void)ws_size;
  const float* adj   = (const float*)d_in[0];   // [1024][128][128]
  const float* feats = (const float*)d_in[1];   // [100000][128]
  const int*   idxs  = (const int*)  d_in[2];   // [1024][128]
  const float* fh    = (const float*)d_in[3];   // [10][128][32]
  const float* ah    = (const float*)d_in[4];   // [45][32]
  float* out = (float*)d_out;
  float* ws  = (float*)d_ws;                    // needs ~832 KB

  rw_pre<<<1, 256, 0, stream>>>(fh, ah, ws);
  rw_main<<<B_BATCH, 256, 0, stream>>>(adj, feats, idxs, ws + WS_ZF, out);
}